// MIL_Loss_47871705481277
// MI455X (gfx1250) — compile-verified
//
#include <hip/hip_runtime.h>
#include <math.h>

typedef float v2f __attribute__((ext_vector_type(2)));
typedef float v8f __attribute__((ext_vector_type(8)));

#define BAG 20
#define IT_BITS 4
#define P1_ITERS (1 << IT_BITS)         // 16 blocks of 64 per wave
#define P1_WG 128
#define P1_TPB 256
#define P1_WAVES (P1_WG * P1_TPB / 32)  // 1024 waves (wid = 10 bits)
#define PART_STRIDE 32
#define P2_TPB 256

// ---------------- Phase 1: pure streaming + f32 WMMA ----------------
// Virtual array v[i] = (i==0)?0:w[i-1], i in [0, 2^20).
// Element index bits:  0-3 = m (A-row), 4-5 = k (A-col), 6-9 = it, 10-19 = wid.
//   A[m][k] = v[wid*1024 + it*64 + 16k + m]
//   B cols: 0 -> ones (per-m sums: bits 0-3 + wave total)
//           1 -> bit0(k) (t4), 2 -> bit1(k) (t5)
//           3+q -> bit_q(it), q=0..3 (t6..t9), updated per iteration
// Bits 10-19 are constant per wave -> recovered in phase 2 from wave totals.
__global__ __launch_bounds__(P1_TPB)
void mil_phase1(const float* __restrict__ w, float* __restrict__ part) {
  const int tid  = blockIdx.x * blockDim.x + threadIdx.x;
  const int wid  = tid >> 5;
  const int lane = threadIdx.x & 31;
  const int n    = lane & 15;           // A row m / B-D column n
  const int klo  = (lane < 16) ? 0 : 2;
  const int hi   = (lane >= 16) ? 1 : 0;

  // Static B entries (cols 0..2); cols 3..6 are per-iteration.
  const float bx0 = (n == 0) ? 1.0f : ((n == 2) ? (float)(klo >> 1) : 0.0f);
  const float by0 = (n <= 1) ? 1.0f : ((n == 2) ? (float)(klo >> 1) : 0.0f);
  const bool  dyn = (n >= 3) && (n <= 2 + IT_BITS);
  const int   sh  = dyn ? (n - 3) : 0;

  const int laneOff  = klo * 16 + n;
  const int waveBase = wid * (P1_ITERS * 64);

  v8f c;
  #pragma unroll
  for (int r = 0; r < 8; ++r) c[r] = 0.0f;

  #pragma unroll 4
  for (int it = 0; it < P1_ITERS; ++it) {
    const int idx0 = waveBase + it * 64 + laneOff;   // virtual index (k = klo)
    const int ld0  = (idx0 > 0) ? (idx0 - 1) : 0;    // always-valid address
    float a0 = w[ld0];
    a0 = (idx0 == 0) ? 0.0f : a0;                    // branchless v_cndmask
    const float a1 = w[idx0 + 15];                   // virtual idx0+16, always >= 16
    v2f a; a.x = a0; a.y = a1;

    const float bv = (float)((it >> sh) & 1);
    v2f b;
    b.x = dyn ? bv : bx0;
    b.y = dyn ? bv : by0;

    c = __builtin_amdgcn_wmma_f32_16x16x4_f32(false, a, false, b,
                                              (short)0, c, false, false);
  }

  // Per-wave partials, fixed slots (deterministic, no float atomics):
  //  slots 0..15  : P_m (column 0, rows hi*8..hi*8+7)
  //  slots 16+2(n-1)+hi, n=1..6 : column-n sums (t4,t5,t6..t9 partials)
  float* pw = part + wid * PART_STRIDE;
  if (n == 0) {
    #pragma unroll
    for (int r = 0; r < 8; ++r) pw[r + hi * 8] = c[r];
  } else if (n <= 2 + IT_BITS) {
    float s = 0.0f;
    #pragma unroll
    for (int r = 0; r < 8; ++r) s += c[r];
    pw[16 + (n - 1) * 2 + hi] = s;
  }
}

// ---------------- Phase 2: tiny finish (single workgroup) ----------------
__global__ __launch_bounds__(P2_TPB)
void mil_phase2(const float* __restrict__ y, const float* __restrict__ part,
                float* __restrict__ out) {
  __shared__ float colred[32][8];
  __shared__ float hred[8][10];

  const int t    = threadIdx.x;
  const int lane = t & 31;
  const int wv   = t >> 5;

  // Column sums over the 1024 wave records (cols 0..27 used).
  const int j  = t & 31;
  const int g8 = t >> 5;
  float cs = 0.0f;
  for (int p = g8; p < P1_WAVES; p += 8) cs += part[p * PART_STRIDE + j];
  colred[j][g8] = cs;

  // High bits: t_{10+q} = sum_p W_p * bit_q(p), W_p = per-wave total.
  float hacc[10];
  #pragma unroll
  for (int q = 0; q < 10; ++q) hacc[q] = 0.0f;
  for (int p = t; p < P1_WAVES; p += P2_TPB) {
    float W = 0.0f;
    #pragma unroll
    for (int mm = 0; mm < 16; ++mm) W += part[p * PART_STRIDE + mm];
    #pragma unroll
    for (int q = 0; q < 10; ++q) hacc[q] += W * (float)((p >> q) & 1);
  }
  #pragma unroll
  for (int q = 0; q < 10; ++q) {
    #pragma unroll
    for (int off = 16; off > 0; off >>= 1) hacc[q] += __shfl_xor(hacc[q], off, 32);
  }
  if (lane == 0) {
    #pragma unroll
    for (int q = 0; q < 10; ++q) hred[wv][q] = hacc[q];
  }
  __syncthreads();

  if (t == 0) {
    float colsum[28];
    for (int jj = 0; jj < 28; ++jj) {
      float s = 0.0f;
      for (int g = 0; g < 8; ++g) s += colred[jj][g];
      colsum[jj] = s;
    }
    float tk[20];
    float T = 0.0f;
    for (int mm = 0; mm < 16; ++mm) T += colsum[mm];
    for (int k = 0; k < 4; ++k) {
      float s = 0.0f;
      for (int mm = 0; mm < 16; ++mm) s += colsum[mm] * (float)((mm >> k) & 1);
      tk[k] = s;
    }
    for (int q = 0; q < 2 + IT_BITS; ++q)          // t4,t5,t6..t9
      tk[4 + q] = colsum[16 + 2 * q] + colsum[17 + 2 * q];
    for (int q = 0; q < 10; ++q) {                  // t10..t19
      float s = 0.0f;
      for (int wq = 0; wq < 8; ++wq) s += hred[wq][q];
      tk[10 + q] = s;
    }
    // result = T*S0 + sum_k d[19-k] * t_k
    float S0 = 0.0f, d[BAG];
    for (int jj = 0; jj < BAG; ++jj) {
      const float p  = y[jj];
      const float lp = logf(p);
      const float l1 = logf(1.0f - p);
      S0 += l1;
      d[jj] = lp - l1;
    }
    float ans = T * S0;
    for (int k = 0; k < 20; ++k) ans += d[19 - k] * tk[k];
    out[0] = ans;
  }
}

extern "C" void kernel_launch(void* const* d_in, const int* in_sizes, int n_in,
                              void* d_out, int out_size, void* d_ws, size_t ws_size,
                              hipStream_t stream) {
  const float* y = (const float*)d_in[0];   // y_prob_list [20]
  const float* w = (const float*)d_in[1];   // w [2^20 - 1]
  float* out  = (float*)d_out;              // scalar f32
  float* part = (float*)d_ws;               // [1024 * 32] per-wave partials

  mil_phase1<<<P1_WG, P1_TPB, 0, stream>>>(w, part);
  mil_phase2<<<1, P2_TPB, 0, stream>>>(y, part, out);
}